// SymbolicReasoningModule_61375082660195
// MI455X (gfx1250) — compile-verified
//
#include <hip/hip_runtime.h>
#include <cstdint>
#include <cstddef>

// Rule-gated row rewrite: out[r,:] = any_rule ? onehot(c1,c2,c3) : x[r,:]
// B = 2,000,000 rows, F = 128 floats/row. Zero-FLOP streaming op; HBM-bound.
//
// Roofline: naive traffic 2.05 GB (~88us @ 23.3 TB/s). P(any_rule) ~= 0.552
// under U[0,1), and one-hot rows don't need columns 3..127, so a 12B header
// probe + conditional full-row read cuts expected traffic to ~1.51 GB (~65us).
//
// Mapping: one wave32 per row (lane l owns a b128 = cols 4l..4l+3, perfectly
// coalesced). Header broadcast via v_readlane -> SALU float compares ->
// scalar branch (no divergence). Pass-through rows are copied via the CDNA5
// async global->LDS->global DMA path (ASYNCcnt), NT-hinted since the 2 GB
// stream is 10x the 192 MB L2.

typedef float v4f __attribute__((ext_vector_type(4)));
typedef int   vi4 __attribute__((vector_size(16)));   // matches async builtin param type

#define AS1 __attribute__((address_space(1)))
#define AS3 __attribute__((address_space(3)))

#ifndef __has_builtin
#define __has_builtin(x) 0
#endif

#if __has_builtin(__builtin_amdgcn_global_load_async_to_lds_b128) && \
    __has_builtin(__builtin_amdgcn_global_store_async_from_lds_b128)
#define USE_ASYNC_LDS 1
#else
#define USE_ASYNC_LDS 0
#endif

// gfx1250 CPol immediate: TH in bits [2:0]; TH=1 -> non-temporal.
#define CPOL_NT 1

__device__ __forceinline__ void wait_asynccnt0() {
#if __has_builtin(__builtin_amdgcn_s_wait_asynccnt)
  __builtin_amdgcn_s_wait_asynccnt(0);
#else
  asm volatile("s_wait_asynccnt 0" ::: "memory");
#endif
}

__global__ __launch_bounds__(256) void symreason_rules_kernel(
    const float* __restrict__ x, float* __restrict__ out, int nrows) {
  // 8 waves per block, one 512B row slot per wave for the async LDS copy path
  __shared__ float lds[8 * 128];

  const int lane = threadIdx.x & 31;
  const int wib  = threadIdx.x >> 5;                 // wave-in-block 0..7
  const int row  = (int)blockIdx.x * 8 + wib;
  if (row >= nrows) return;

  const size_t base = (size_t)row * 128;
  const float* __restrict__ rowp = x + base;
  float* __restrict__ outp = out + base;

  // --- 12B header: lanes 0..2 fetch cols 0..2 (hunger,social,wealth). RT
  // caching on purpose: the same cacheline may be re-read by the copy path.
  float hv = 0.0f;
  if (lane < 3) hv = rowp[lane];

  // Broadcast to SGPRs -> rule evaluation is wave-uniform scalar work
  const float hunger = __int_as_float(__builtin_amdgcn_readlane(__float_as_int(hv), 0));
  const float social = __int_as_float(__builtin_amdgcn_readlane(__float_as_int(hv), 1));
  const float wealth = __int_as_float(__builtin_amdgcn_readlane(__float_as_int(hv), 2));

  const bool c1 = hunger > 0.8f;
  const bool c2 = (!c1) && (social < 0.3f);
  const bool c3 = (!c1) && (!c2) && (wealth < 0.2f);
  const int  any = (int)(c1 || c2 || c3);

  if (__builtin_amdgcn_readfirstlane(any)) {
    // One-hot row: columns 3..127 of the input are never read.
    v4f v = {0.0f, 0.0f, 0.0f, 0.0f};
    if (lane == 0) {
      v.x = c1 ? 1.0f : 0.0f;
      v.y = c2 ? 1.0f : 0.0f;
      v.z = c3 ? 1.0f : 0.0f;
    }
    __builtin_nontemporal_store(v, (v4f*)outp + lane);   // 512B/row, streaming
  } else {
    // Pass-through: copy 512B row.
#if USE_ASYNC_LDS
    // CDNA5 async DMA path: global -> LDS -> global, no VGPR staging. NT on
    // both legs: each byte is touched exactly once.
    float* slot = &lds[wib * 128 + lane * 4];
    AS3 vi4* lslot = (AS3 vi4*)slot;
    AS1 vi4* gsrc  = (AS1 vi4*)((const v4f*)rowp + lane);
    AS1 vi4* gdst  = (AS1 vi4*)((v4f*)outp + lane);
    __builtin_amdgcn_global_load_async_to_lds_b128(gsrc, lslot, 0, CPOL_NT);
    wait_asynccnt0();  // LDS slot written before the dependent async store reads it
    __builtin_amdgcn_global_store_async_from_lds_b128(gdst, lslot, 0, CPOL_NT);
    // store completion covered by implicit wait-idle at S_ENDPGM
#else
    v4f v = __builtin_nontemporal_load((const v4f*)rowp + lane);
    __builtin_nontemporal_store(v, (v4f*)outp + lane);
#endif
  }
}

extern "C" void kernel_launch(void* const* d_in, const int* in_sizes, int n_in,
                              void* d_out, int out_size, void* d_ws, size_t ws_size,
                              hipStream_t stream) {
  (void)n_in; (void)d_ws; (void)ws_size; (void)out_size;
  const float* x  = (const float*)d_in[0];
  float* out      = (float*)d_out;
  const int nrows = in_sizes[0] / 128;      // 2,000,000
  const int rows_per_block = 8;             // 256 threads = 8 wave32
  dim3 grid((unsigned)((nrows + rows_per_block - 1) / rows_per_block));
  symreason_rules_kernel<<<grid, dim3(256), 0, stream>>>(x, out, nrows);
}